// PointPillarScatter_7035156431557
// MI455X (gfx1250) — compile-verified
//
#include <hip/hip_runtime.h>

// PointPillarScatter for MI455X (gfx1250).
// Pure data-movement op (0 FLOPs): roofline floor ~= 290 MB / 23.3 TB/s ~= 12.5 us.
// Strategy: invert the scatter into a gather so HBM traffic is fully coalesced
// 128-bit streaming stores, and all random accesses become reads against an
// L2-resident working set (25.6 MB features + 4 MB pid grid << 192 MB L2).
// Non-temporal store hints keep the 256 MB output stream from evicting that
// working set; global_prefetch_b8 covers the pid stream.

typedef float v4f __attribute__((ext_vector_type(4)));
typedef int   v4i __attribute__((ext_vector_type(4)));

constexpr int C_    = 64;
constexpr int NX_   = 1024;
constexpr int NY_   = 1024;
constexpr int P_    = 100000;
constexpr int NCELL = NX_ * NY_;      // 1,048,576 BEV cells

// ---------------- Pass 1a: pid grid := -1 (kept Regular-Temporal -> L2 resident)
__global__ void pps_pid_init(int* __restrict__ pid) {
    int i = blockIdx.x * blockDim.x + threadIdx.x;          // int4 index
    v4i m1 = { -1, -1, -1, -1 };
    reinterpret_cast<v4i*>(pid)[i] = m1;
}

// ---------------- Pass 1b: scatter pillar ids (4-byte scatter into 4 MB grid; L2 absorbs it)
__global__ void pps_pid_scatter(const int* __restrict__ coords, int* __restrict__ pid) {
    int p = blockIdx.x * blockDim.x + threadIdx.x;
    if (p >= P_) return;
    int z = coords[3 * p + 0];
    int y = coords[3 * p + 1];
    int x = coords[3 * p + 2];
    pid[z + y * NX_ + x] = p;
}

// ---------------- Pass 2: coalesced gather-write of the dense BEV grid
// grid = (256, 64); block = 256. Each thread grid-strides over float4 cells of
// one channel row. pid loads + feature gathers hit L2; output stores are NT b128.
__global__ void pps_gather(const float* __restrict__ feat,
                           const int*   __restrict__ pid,
                           float*       __restrict__ out) {
    const int c = blockIdx.y;
    const float* __restrict__ frow = feat + (size_t)c * P_;
    v4f* __restrict__ orow = reinterpret_cast<v4f*>(out + (size_t)c * NCELL);
    const v4i* __restrict__ pid4 = reinterpret_cast<const v4i*>(pid);

    const int n4     = NCELL / 4;                 // 262,144 float4 per channel
    const int stride = gridDim.x * blockDim.x;    // 65,536 threads per channel
    for (int i = blockIdx.x * blockDim.x + threadIdx.x; i < n4; i += stride) {
        if (i + stride < n4)
            __builtin_prefetch(&pid4[i + stride], 0, 1);   // -> global_prefetch_b8
        v4i p4 = pid4[i];
        v4f v;
        v.x = (p4.x >= 0) ? frow[p4.x] : 0.0f;
        v.y = (p4.y >= 0) ? frow[p4.y] : 0.0f;
        v.z = (p4.z >= 0) ? frow[p4.z] : 0.0f;
        v.w = (p4.w >= 0) ? frow[p4.w] : 0.0f;
        __builtin_nontemporal_store(v, &orow[i]);          // b128 store, TH=NT
    }
}

// ---------------- Fallback (only if workspace < 4 MB): zero-fill + direct scatter
__global__ void pps_zero(float* __restrict__ out) {
    size_t i = (size_t)blockIdx.x * blockDim.x + threadIdx.x;  // float4 index
    v4f z = { 0.0f, 0.0f, 0.0f, 0.0f };
    __builtin_nontemporal_store(z, &reinterpret_cast<v4f*>(out)[i]);
}

__global__ void pps_scatter_direct(const float* __restrict__ feat,
                                   const int*   __restrict__ coords,
                                   float*       __restrict__ out) {
    int p = blockIdx.x * blockDim.x + threadIdx.x;
    if (p >= P_) return;
    int idx = coords[3 * p + 0] + coords[3 * p + 1] * NX_ + coords[3 * p + 2];
#pragma unroll 4
    for (int c = 0; c < C_; ++c) {
        float v = feat[(size_t)c * P_ + p];                    // coalesced read
        __builtin_nontemporal_store(v, &out[(size_t)c * NCELL + idx]);
    }
}

extern "C" void kernel_launch(void* const* d_in, const int* in_sizes, int n_in,
                              void* d_out, int out_size, void* d_ws, size_t ws_size,
                              hipStream_t stream) {
    (void)in_sizes; (void)n_in; (void)out_size;
    const float* feat   = (const float*)d_in[0];   // [C, P] f32
    const int*   coords = (const int*)d_in[1];     // [1, P, 3] i32 (z, y, x)
    float*       out    = (float*)d_out;           // [1, C, NY, NX] f32

    if (ws_size >= (size_t)NCELL * sizeof(int)) {
        int* pid = (int*)d_ws;
        pps_pid_init<<<NCELL / 4 / 256, 256, 0, stream>>>(pid);
        pps_pid_scatter<<<(P_ + 255) / 256, 256, 0, stream>>>(coords, pid);
        dim3 g(256, C_);
        pps_gather<<<g, 256, 0, stream>>>(feat, pid, out);
    } else {
        pps_zero<<<((size_t)NCELL * C_ / 4) / 256, 256, 0, stream>>>(out);
        pps_scatter_direct<<<(P_ + 255) / 256, 256, 0, stream>>>(feat, coords, out);
    }
}